// GatingNetworkLayer_41283225649157
// MI455X (gfx1250) — compile-verified
//
#include <hip/hip_runtime.h>
#include <math.h>

typedef __attribute__((ext_vector_type(2))) float v2f;
typedef __attribute__((ext_vector_type(8))) float v8f;

#define BB    32
#define TT    512
#define KK    8
#define SS    8
#define DD    300
#define CC    511   // T-1
#define CPAD  512   // padded context axis (K-dim for m GEMM)
#define ATS   512   // aT row stride

// ---------------------------------------------------------------------------
// 1) v[b,c,:] = mean over S of ctx[b,c,s,:]   (the bandwidth-dominant pass)
// ---------------------------------------------------------------------------
__global__ __launch_bounds__(256) void mean_kernel(const float* __restrict__ ctx,
                                                   float* __restrict__ v) {
    int b = blockIdx.y;
    int t = blockIdx.x * 256 + threadIdx.x;
    if (t >= CC * DD) return;
    int c  = t / DD;
    int dd = t - c * DD;
    const float* p = ctx + (((size_t)b * CC + c) * SS) * DD + dd;
    float s = 0.f;
#pragma unroll
    for (int j = 0; j < SS; ++j) s += p[(size_t)j * DD];
    v[((size_t)b * CPAD + c) * DD + dd] = s * (1.0f / SS);
}

// ---------------------------------------------------------------------------
// 2) aT[b][k][c] = (center[b,k,:] . v[b,c,:]) / sqrt(d)   via V_WMMA_F32_16X16X4_F32
//    One wave per (b, 16-wide c tile). A-rows 8..15 load duplicate data (row n&7):
//    they only pollute D-rows 8..15, which are never consumed. Branchless loop.
// ---------------------------------------------------------------------------
__global__ __launch_bounds__(32) void a_wmma_kernel(const float* __restrict__ center,
                                                    const float* __restrict__ v,
                                                    float* __restrict__ aT) {
    int b    = blockIdx.y;
    int c0   = blockIdx.x * 16;
    int lane = threadIdx.x;
    int half = lane >> 4;      // which K-pair this lane holds
    int n    = lane & 15;      // A: M row; B/D: N column
    const float* cen = center + ((size_t)b * KK + (n & 7)) * DD;  // dup rows 8..15
    const float* vb  = v + ((size_t)b * CPAD + (c0 + n)) * DD;    // c==511 pad row ok
    v8f acc = {};
#pragma unroll 5
    for (int dd = 0; dd < DD; dd += 4) {
        int ka = dd + 2 * half;                           // K indices {ka, ka+1}
        v2f a  = *(const v2f*)(cen + ka);
        v2f bb = *(const v2f*)(vb + ka);                  // B rows ka,ka+1 at col c
        acc = __builtin_amdgcn_wmma_f32_16x16x4_f32(false, a, false, bb,
                                                    (short)0, acc, false, false);
    }
    const float scale = 0.05773502691896258f;  // 1/sqrt(300)
    float* out = aT + (size_t)b * 16 * ATS + (c0 + n);
#pragma unroll
    for (int r = 0; r < 8; ++r)
        out[(size_t)(r + 8 * half) * ATS] = acc[r] * scale;
}

// ---------------------------------------------------------------------------
// 3) In-place softmax over c for each (b,k<8) row of aT, with positional
//    log-weight. Zeroes the c=511 pad column so the m-GEMM K-padding is exact.
// ---------------------------------------------------------------------------
__device__ __forceinline__ float logw_at(int c, int cp) {
    int ctx = c + (c >= cp);
    float dist = fabsf((float)(ctx - cp));
    float w = expf(-dist * 1000.0f);           // SIGMA = 0.001
    return logf(fmaxf(w, 1e-12f));             // EPS_LOG
}

__global__ __launch_bounds__(256) void softmax_kernel(const int* __restrict__ cpos,
                                                      float* __restrict__ aT) {
    int k = blockIdx.x, b = blockIdx.y, tid = threadIdx.x;
    int cp = cpos[b];
    float* row = aT + ((size_t)b * 16 + k) * ATS;
    __shared__ float red[256];
    bool v1 = (tid + 256) < CC;
    float x0 = row[tid] + logw_at(tid, cp);
    float x1 = v1 ? (row[tid + 256] + logw_at(tid + 256, cp)) : -INFINITY;

    red[tid] = fmaxf(x0, x1); __syncthreads();
    for (int s = 128; s > 0; s >>= 1) { if (tid < s) red[tid] = fmaxf(red[tid], red[tid + s]); __syncthreads(); }
    float M = red[0]; __syncthreads();

    float e0 = expf(x0 - M);
    float e1 = v1 ? expf(x1 - M) : 0.f;
    red[tid] = e0 + e1; __syncthreads();
    for (int s = 128; s > 0; s >>= 1) { if (tid < s) red[tid] += red[tid + s]; __syncthreads(); }
    float inv = 1.0f / red[0];

    row[tid] = e0 * inv;
    if (v1) row[tid + 256] = e1 * inv;
    if (tid == 255) row[CC] = 0.0f;            // kill pad column c=511
}

// ---------------------------------------------------------------------------
// 4) m[b][k][d] = sum_c alphaT[b][k][c] * v[b][c][d]   via V_WMMA_F32_16X16X4_F32
//    One wave per (b, 16-wide d tile); K-dim = padded c axis (512). B columns
//    >= 300 load clamped addresses: they only pollute D-cols >= 300, which the
//    guarded store skips. Branchless inner loop.
// ---------------------------------------------------------------------------
__global__ __launch_bounds__(32) void m_wmma_kernel(const float* __restrict__ aT,
                                                    const float* __restrict__ v,
                                                    float* __restrict__ mbuf) {
    int b    = blockIdx.y;
    int d0   = blockIdx.x * 16;
    int lane = threadIdx.x;
    int half = lane >> 4;
    int n    = lane & 15;
    const float* al = aT + ((size_t)b * 16 + n) * ATS;   // A row = sense index n
    int col  = d0 + n;
    int colc = col < DD ? col : (DD - 1);                // clamped B column
    const float* vb = v + (size_t)b * CPAD * DD + colc;
    v8f acc = {};
#pragma unroll 4
    for (int cc = 0; cc < CPAD; cc += 4) {
        int ca = cc + 2 * half;                          // c indices {ca, ca+1}
        v2f a = *(const v2f*)(al + ca);                  // alphaT[n][ca..ca+1]
        v2f bb;
        bb.x = vb[(size_t)ca * DD];                      // v[ca][colc]
        bb.y = vb[(size_t)(ca + 1) * DD];                // v[ca+1][colc]
        acc = __builtin_amdgcn_wmma_f32_16x16x4_f32(false, a, false, bb,
                                                    (short)0, acc, false, false);
    }
    if (col < DD) {
        float* out = mbuf + (size_t)b * 16 * DD + col;
#pragma unroll
        for (int r = 0; r < 8; ++r)
            out[(size_t)(r + 8 * half) * DD] = acc[r];
    }
}

// ---------------------------------------------------------------------------
// 5) cosine sim over d, softmax over K, argmax, pooled gather.
//    out = [ pooled (32x300) | q (32x8) ]
// ---------------------------------------------------------------------------
__global__ __launch_bounds__(256) void final_kernel(const float* __restrict__ center,
                                                    const float* __restrict__ mbuf,
                                                    float* __restrict__ out) {
    int b = blockIdx.x, tid = threadIdx.x;
    int warp = tid >> 5, lane = tid & 31;
    __shared__ float sarr[8];
    __shared__ float sq[8];
    __shared__ int   sidx;
    if (warp < KK) {
        const float* cen = center + ((size_t)b * KK + warp) * DD;
        const float* mm  = mbuf + ((size_t)b * 16 + warp) * DD;
        float num = 0.f, n1 = 0.f, n2 = 0.f;
        for (int d = lane; d < DD; d += 32) {
            float ce = cen[d], mv = mm[d];
            num += ce * mv; n1 += ce * ce; n2 += mv * mv;
        }
#pragma unroll
        for (int off = 16; off > 0; off >>= 1) {
            num += __shfl_down(num, off, 32);
            n1  += __shfl_down(n1,  off, 32);
            n2  += __shfl_down(n2,  off, 32);
        }
        if (lane == 0) {
            float d1 = fmaxf(sqrtf(n1), 1e-8f);   // EPS_COS
            float d2 = fmaxf(sqrtf(n2), 1e-8f);
            sarr[warp] = num / (d1 * d2);
        }
    }
    __syncthreads();
    if (tid == 0) {
        float mx = sarr[0];
        for (int j = 1; j < KK; ++j) mx = fmaxf(mx, sarr[j]);
        float e[KK]; float sum = 0.f;
        for (int j = 0; j < KK; ++j) { e[j] = expf(sarr[j] - mx); sum += e[j]; }
        float inv = 1.f / sum;
        int bi = 0; float bq = e[0] * inv; sq[0] = bq;
        for (int j = 1; j < KK; ++j) {
            float qj = e[j] * inv; sq[j] = qj;
            if (qj > bq) { bq = qj; bi = j; }   // first-max tie-break like argmax
        }
        sidx = bi;
    }
    __syncthreads();
    if (tid < KK) out[(size_t)BB * DD + (size_t)b * KK + tid] = sq[tid];
    int idx = sidx;
    const float* src = center + ((size_t)b * KK + idx) * DD;
    for (int d = tid; d < DD; d += 256) out[(size_t)b * DD + d] = src[d];
}

// ---------------------------------------------------------------------------
extern "C" void kernel_launch(void* const* d_in, const int* in_sizes, int n_in,
                              void* d_out, int out_size, void* d_ws, size_t ws_size,
                              hipStream_t stream) {
    (void)in_sizes; (void)n_in; (void)out_size; (void)ws_size;
    const int*   cpos   = (const int*)d_in[0];
    // d_in[1] (query_token_ids) only defines shapes; unused in math.
    const float* center = (const float*)d_in[2];
    const float* ctx    = (const float*)d_in[3];
    float* out = (float*)d_out;

    float* v    = (float*)d_ws;                        // [B][512][300]
    float* aT   = v  + (size_t)BB * CPAD * DD;         // [B][16][512]
    float* mbuf = aT + (size_t)BB * 16 * ATS;          // [B][16][300]

    dim3 g1((CC * DD + 255) / 256, BB);
    mean_kernel<<<g1, 256, 0, stream>>>(ctx, v);

    dim3 g2(CPAD / 16, BB);
    a_wmma_kernel<<<g2, 32, 0, stream>>>(center, v, aT);

    dim3 g3(KK, BB);
    softmax_kernel<<<g3, 256, 0, stream>>>(cpos, aT);

    dim3 g4((DD + 15) / 16, BB);
    m_wmma_kernel<<<g4, 32, 0, stream>>>(aT, v, mbuf);

    final_kernel<<<BB, 256, 0, stream>>>(center, mbuf, out);
}